// SpanScorerCRF_85392539779541
// MI455X (gfx1250) — compile-verified
//
#include <hip/hip_runtime.h>
#include <hip/hip_bf16.h>

typedef float v2f __attribute__((ext_vector_type(2)));
typedef float v8f __attribute__((ext_vector_type(8)));

#define B_DIM 32
#define L_DIM 512
#define D_DIM 1024
#define T_SEQ 15
#define N_POS 7
#define NUM_TAGS 8

// ---------------------------------------------------------------------------
// Kernel 1: seq_scores[b,l,t] = sum_d seq[b,l,d] * W[d,t] + bias[t]
// One wave (32 threads) per 16-row output tile, fp32 WMMA 16x16x4.
// ---------------------------------------------------------------------------
__global__ __launch_bounds__(32)
void spanscorer_gemm_wmma(const float* __restrict__ A,   // [B*L, D]
                          const float* __restrict__ W,   // [D, 15]
                          const float* __restrict__ bias,// [15]
                          float* __restrict__ out)       // [B*L, 15]
{
    const int lane = threadIdx.x;          // 0..31
    const int half = lane >> 4;            // 0: K 0/1, 1: K 2/3
    const int lid  = lane & 15;            // M for A, N for B/C
    const long rowBase = (long)blockIdx.x * 16;

    // A: lane holds row (rowBase+lid), columns (k + half*2 + {0,1})
    const float* arow = A + (rowBase + lid) * D_DIM + half * 2;

    // B column index, clamped so OOB lane 15 reads a valid address
    const int n = (lid < T_SEQ) ? lid : 0;
    const bool ncol = (lid < T_SEQ);

    v8f acc = {0.f, 0.f, 0.f, 0.f, 0.f, 0.f, 0.f, 0.f};

    for (int k = 0; k < D_DIM; k += 4) {
        v2f a;
        a.x = arow[k + 0];
        a.y = arow[k + 1];

        const int kr = k + half * 2;       // <= 1022
        float w0 = W[(long)(kr + 0) * T_SEQ + n];
        float w1 = W[(long)(kr + 1) * T_SEQ + n];
        v2f bm;
        bm.x = ncol ? w0 : 0.f;            // value select, no EXEC change
        bm.y = ncol ? w1 : 0.f;

        acc = __builtin_amdgcn_wmma_f32_16x16x4_f32(
            /*neg_a=*/false, a, /*neg_b=*/false, bm,
            /*c_mod=*/(short)0, acc, /*reuse_a=*/false, /*reuse_b=*/false);
    }

    // C/D layout: VGPR r -> M = r + half*8, N = lid
    if (ncol) {
        const float bb = bias[n];
#pragma unroll
        for (int r = 0; r < 8; ++r) {
            const long m = rowBase + r + half * 8;
            out[m * T_SEQ + lid] = acc[r] + bb;
        }
    }
}

// ---------------------------------------------------------------------------
// Kernel 2: Viterbi decode. One wave per batch element; lane j owns tag j.
// Backpointers in LDS, lane 0 backtraces, path (bytes) written to workspace.
// ---------------------------------------------------------------------------
__global__ __launch_bounds__(32)
void spanscorer_viterbi(const float* __restrict__ scores,    // [B, L, 15]
                        const unsigned char* __restrict__ mask, // [B, L] bool
                        const float* __restrict__ trans,     // [15, 15]
                        const float* __restrict__ start_t,   // [15]
                        const float* __restrict__ end_t,     // [15]
                        unsigned char* __restrict__ path)    // ws: [B, L]
{
    __shared__ unsigned char bp[L_DIM * T_SEQ];   // bp[t*15 + j], t>=1

    const int b = blockIdx.x;
    const int j = threadIdx.x;
    const int jj = (j < T_SEQ) ? j : 0;
    const bool act = (j < T_SEQ);

    const float* sc = scores + (long)b * L_DIM * T_SEQ;
    const unsigned char* mk = mask + (long)b * L_DIM;

    // transition column trans[i][j] in registers
    float tc[T_SEQ];
#pragma unroll
    for (int i = 0; i < T_SEQ; ++i) tc[i] = trans[i * T_SEQ + jj];

    const float NEG = -3.0e38f;
    float alpha = act ? (start_t[jj] + sc[jj]) : NEG;

    for (int t = 1; t < L_DIM; ++t) {
        const float lg = act ? sc[t * T_SEQ + jj] : 0.f;
        float best = NEG;
        int bi = 0;
#pragma unroll
        for (int i = 0; i < T_SEQ; ++i) {
            const float ai = __shfl(alpha, i, 32);
            const float v = (ai + tc[i]) + lg;   // match JAX op order
            const bool gt = v > best;            // strict: first max wins
            best = gt ? v : best;
            bi   = gt ? i : bi;
        }
        const bool m = (mk[t] != 0);
        alpha = m ? best : alpha;
        bi    = m ? bi : j;
        if (act) bp[t * T_SEQ + j] = (unsigned char)bi;
    }

    const float fin = act ? (alpha + end_t[jj]) : NEG;
    __syncthreads();

    if (j == 0) {
        // argmax over final alphas (first index wins)
        int last = 0;
        float bb = NEG;
#pragma unroll
        for (int i = 0; i < T_SEQ; ++i) {
            const float v = __shfl(fin, i, 32);
            if (v > bb) { bb = v; last = i; }
        }
        unsigned char* pb = path + (long)b * L_DIM;
        int tag = last;
        pb[L_DIM - 1] = (unsigned char)tag;
        for (int t = L_DIM - 1; t >= 1; --t) {
            tag = bp[t * T_SEQ + tag];
            pb[t - 1] = (unsigned char)tag;
        }
    }
}

// ---------------------------------------------------------------------------
// Kernel 3: span labeling + one-hot. One thread per (batch, span).
// ---------------------------------------------------------------------------
__global__ void spanscorer_spans(const int* __restrict__ span_map, // [S,2]
                                 const unsigned char* __restrict__ path, // [B,L]
                                 float* __restrict__ out,          // [B,S,8]
                                 int S)
{
    const int idx = blockIdx.x * blockDim.x + threadIdx.x;
    const int total = B_DIM * S;
    if (idx >= total) return;
    const int b  = idx / S;
    const int sp = idx % S;

    const int s = span_map[sp * 2 + 0];
    const int e = span_map[sp * 2 + 1];
    const unsigned char* pb = path + (long)b * L_DIM;

    const int tag_s = pb[s];
    const bool is_B = (tag_s >= 1) && (tag_s <= N_POS);
    const int I_id = tag_s + N_POS;

    bool interior_ok = true;
    for (int p = s + 1; p < e; ++p)
        interior_ok = interior_ok && (pb[p] == I_id);

    const int e_clip = (e < L_DIM - 1) ? e : (L_DIM - 1);
    const bool next_ok = (e >= L_DIM) || (pb[e_clip] != I_id);

    const int lab = (is_B && interior_ok && next_ok) ? tag_s : 0;

    float* o = out + (long)idx * NUM_TAGS;
#pragma unroll
    for (int k = 0; k < NUM_TAGS; ++k) o[k] = (k == lab) ? 1.f : 0.f;
}

// ---------------------------------------------------------------------------
extern "C" void kernel_launch(void* const* d_in, const int* in_sizes, int n_in,
                              void* d_out, int out_size, void* d_ws, size_t ws_size,
                              hipStream_t stream) {
    const float*         seq      = (const float*)d_in[0];         // [B,L,D]
    const unsigned char* mask     = (const unsigned char*)d_in[1]; // [B,L] bool
    const int*           span_map = (const int*)d_in[2];           // [S,2]
    const float*         W        = (const float*)d_in[3];         // [D,15]
    const float*         bias     = (const float*)d_in[4];         // [15]
    const float*         trans    = (const float*)d_in[5];         // [15,15]
    const float*         start_t  = (const float*)d_in[6];         // [15]
    const float*         end_t    = (const float*)d_in[7];         // [15]

    float* out = (float*)d_out;
    const int S = in_sizes[2] / 2;   // 4068

    float* seq_scores = out;                                  // [B,L,15]
    float* span_pred  = out + (long)B_DIM * L_DIM * T_SEQ;    // [B,S,8]
    unsigned char* path = (unsigned char*)d_ws;               // [B,L]

    // 1) GEMM via fp32 WMMA: 1024 tiles of 16 rows
    spanscorer_gemm_wmma<<<(B_DIM * L_DIM) / 16, 32, 0, stream>>>(
        seq, W, bias, seq_scores);

    // 2) Viterbi: one wave per batch element
    spanscorer_viterbi<<<B_DIM, 32, 0, stream>>>(
        seq_scores, mask, trans, start_t, end_t, path);

    // 3) Span one-hot
    const int total = B_DIM * S;
    spanscorer_spans<<<(total + 255) / 256, 256, 0, stream>>>(
        span_map, path, span_pred, S);
}